// JointsSmoothL1Loss_26912265077283
// MI455X (gfx1250) — compile-verified
//
#include <hip/hip_runtime.h>
#include <hip/hip_bf16.h>
#include <math.h>

// Problem constants from the reference
#define IMG_W 384.0f
#define IMG_H 512.0f
#define BETA 0.05f
#define NB 16384
#define NV 8
#define NJ 32

#define THREADS 256
#define NWAVE 8            // 8 wave32 waves per block
#define PROBS_PER_WAVE 32  // each wave loops over 32 problems -> 256 per block
#define JACOBI_SWEEPS 6

typedef __attribute__((ext_vector_type(2))) float v2f;
typedef __attribute__((ext_vector_type(8))) float v8f;

__global__ void zero_kernel(float* out) {
  if (threadIdx.x == 0) out[0] = 0.0f;
}

// Wave-local LDS ordering: CDNA5 keeps a wave's DS ops in order (DScnt, §7.3),
// so we only need to (a) wait for outstanding DS ops and (b) stop the compiler
// from reordering LDS accesses across this point. No block barrier needed.
__device__ __forceinline__ void wave_lds_fence() {
  asm volatile("s_wait_dscnt 0" ::: "memory");
  __builtin_amdgcn_wave_barrier();
}

__global__ __launch_bounds__(THREADS) void dlt_loss_kernel(
    const float* __restrict__ pred2d,   // (B, V, J, 2)
    const float* __restrict__ gt3d,     // (B, J, 3)
    const float* __restrict__ proj,     // (B, V, 3, 4)
    const float* __restrict__ valid,    // (B, V, J)
    float* __restrict__ out)            // scalar
{
  __shared__ float lds_rows[NWAVE][16][4];   // per-wave weighted DLT rows
  __shared__ float lds_D[NWAVE][8][32];      // per-wave WMMA D dump
  __shared__ float lds_M[THREADS][17];       // 4x4 Gram per problem (pad 17 -> no bank conflicts)
  __shared__ float lds_red[THREADS];

  const int tid   = threadIdx.x;
  const int w     = tid >> 5;
  const int lane  = tid & 31;
  const int pbase = blockIdx.x * THREADS;    // grid sized exactly (B*J)/256

  // ================= Phase 1: Gram matrices via WMMA (1 problem / wave / iter) ====
  const int lm = lane & 15;
  const int hi = lane >> 4;
  const float* R = &lds_rows[w][0][0];       // R[r*4 + c] = row_r[c]

#pragma unroll 1
  for (int t = 0; t < PROBS_PER_WAVE; ++t) {
    const int prob = pbase + w * PROBS_PER_WAVE + t;
    const int b = prob >> 5;                 // NJ == 32
    const int j = prob & 31;

    // lanes 0..15: one weighted DLT row each (row = 2v+s; row order irrelevant for A^T A)
    if (lane < 16) {
      const int v  = lane >> 1;
      const int s  = lane & 1;               // 0 -> (x, P0), 1 -> (y, P1)
      const int bv = b * NV + v;
      const float px = pred2d[(bv * NJ + j) * 2 + 0];
      const float py = pred2d[(bv * NJ + j) * 2 + 1];
      const float tt = s ? (py + 1.0f) * (0.5f * IMG_H)
                         : (px + 1.0f) * (0.5f * IMG_W);
      const float wv = valid[bv * NJ + j];
      const float* Pb = proj + bv * 12;      // rows P0,P1,P2 of the 3x4
#pragma unroll
      for (int c = 0; c < 4; ++c)
        lds_rows[w][lane][c] = wv * (tt * Pb[8 + c] - Pb[s * 4 + c]);
    }
    wave_lds_fence();

    // Diagonal-block trick: A_w rows 4g..4g+3 = R_g^T, B_w = A_w^T; the four
    // diagonal 4x4 blocks of D = A_w*B_w are R_g^T R_g, summing to A^T A.
    // f32 A 16x4 layout: VGPR0 = K0 (lanes 0-15) / K2 (16-31); VGPR1 = K1/K3.
    // f32 B 4x16 layout: VGPR0 = K0 (lanes 0-15) / K1 (16-31); VGPR1 = K2/K3.
    const int g = lm >> 2, c = lm & 3;
    v2f a, bb;
    a.x  = R[(4 * g + 2 * hi    ) * 4 + c];
    a.y  = R[(4 * g + 2 * hi + 1) * 4 + c];
    bb.x = R[(4 * g + hi        ) * 4 + c];
    bb.y = R[(4 * g + 2 + hi    ) * 4 + c];
    v8f acc = {};
    acc = __builtin_amdgcn_wmma_f32_16x16x4_f32(
        false, a, false, bb, (short)0, acc, false, false);

    // Dump D, then lanes 0..15 gather one M element each into compact storage.
#pragma unroll
    for (int r = 0; r < 8; ++r) lds_D[w][r][lane] = acc[r];
    wave_lds_fence();

    if (lane < 16) {
      const int c1 = lane >> 2, c2 = lane & 3;
      // C/D layout: VGPR r, lanes 0-15 -> D[r][n]; lanes 16-31 -> D[r+8][n].
      const float m = lds_D[w][c1][c2]              // block g=0
                    + lds_D[w][4 + c1][4 + c2]      // g=1
                    + lds_D[w][c1][24 + c2]         // g=2 (rows 8..11)
                    + lds_D[w][4 + c1][28 + c2];    // g=3 (rows 12..15)
      lds_M[w * PROBS_PER_WAVE + t][lane] = m;
    }
    wave_lds_fence();
  }

  __syncthreads();

  // ================= Phase 2: one eigensolve per thread (full wave32 occupancy) ===
  const int prob = pbase + tid;
  const int b = prob >> 5;
  const int j = prob & 31;

  float Mm[4][4];
#pragma unroll
  for (int c1 = 0; c1 < 4; ++c1)
#pragma unroll
    for (int c2 = 0; c2 < 4; ++c2)
      Mm[c1][c2] = 0.5f * (lds_M[tid][c1 * 4 + c2] + lds_M[tid][c2 * 4 + c1]);

  float Ev[4][4];
#pragma unroll
  for (int i = 0; i < 4; ++i)
#pragma unroll
    for (int k = 0; k < 4; ++k) Ev[i][k] = (i == k) ? 1.0f : 0.0f;

  // Branchless cyclic Jacobi: rotations always applied; negligible pivots get
  // (cth,sth)=(1,0) via selects -> no divergence, all lanes productive.
#pragma unroll 1
  for (int sweep = 0; sweep < JACOBI_SWEEPS; ++sweep) {
#pragma unroll
    for (int p = 0; p < 3; ++p) {
#pragma unroll
      for (int q = p + 1; q < 4; ++q) {
        const float apq   = Mm[p][q];
        const float scale = fabsf(Mm[p][p]) + fabsf(Mm[q][q]);
        const bool  doRot = fabsf(apq) > (1e-12f * scale + 1e-30f);
        const float denom = doRot ? apq : 1.0f;
        const float theta = 0.5f * (Mm[q][q] - Mm[p][p]) / denom;
        const float tpar  = copysignf(1.0f, theta) /
                            (fabsf(theta) + sqrtf(1.0f + theta * theta));
        float cth = 1.0f / sqrtf(1.0f + tpar * tpar);
        float sth = tpar * cth;
        cth = doRot ? cth : 1.0f;
        sth = doRot ? sth : 0.0f;
#pragma unroll
        for (int k = 0; k < 4; ++k) {          // columns
          const float mkp = Mm[k][p], mkq = Mm[k][q];
          Mm[k][p] = cth * mkp - sth * mkq;
          Mm[k][q] = sth * mkp + cth * mkq;
        }
#pragma unroll
        for (int k = 0; k < 4; ++k) {          // rows
          const float mpk = Mm[p][k], mqk = Mm[q][k];
          Mm[p][k] = cth * mpk - sth * mqk;
          Mm[q][k] = sth * mpk + cth * mqk;
        }
#pragma unroll
        for (int k = 0; k < 4; ++k) {          // eigenvectors
          const float vkp = Ev[k][p], vkq = Ev[k][q];
          Ev[k][p] = cth * vkp - sth * vkq;
          Ev[k][q] = sth * vkp + cth * vkq;
        }
      }
    }
  }

  // Smallest-eigenvalue column (sign ambiguity cancels in X[:3]/X[3])
  int bk = 0; float bvv = Mm[0][0];
  if (Mm[1][1] < bvv) { bvv = Mm[1][1]; bk = 1; }
  if (Mm[2][2] < bvv) { bvv = Mm[2][2]; bk = 2; }
  if (Mm[3][3] < bvv) { bvv = Mm[3][3]; bk = 3; }
  float X[4];
#pragma unroll
  for (int i = 0; i < 4; ++i)
    X[i] = (bk == 0) ? Ev[i][0] : (bk == 1) ? Ev[i][1]
         : (bk == 2) ? Ev[i][2] : Ev[i][3];

  const float inv = 1.0f / X[3];
  float lossp = 0.0f;
#pragma unroll
  for (int d = 0; d < 3; ++d) {
    const float pr = X[d] * inv;
    const float gv = gt3d[(b * NJ + j) * 3 + d];
    const float ad = fabsf(pr - gv);
    lossp += (ad < BETA) ? (0.5f * ad * ad / BETA) : (ad - 0.5f * BETA);
  }

  // ================= Phase 3: block tree-reduction + single atomic ==============
  lds_red[tid] = lossp;
  __syncthreads();
#pragma unroll
  for (int s = THREADS / 2; s > 0; s >>= 1) {
    if (tid < s) lds_red[tid] += lds_red[tid + s];
    __syncthreads();
  }
  if (tid == 0) atomicAdd(out, lds_red[0] * (1.0f / (float)NB));
}

extern "C" void kernel_launch(void* const* d_in, const int* in_sizes, int n_in,
                              void* d_out, int out_size, void* d_ws, size_t ws_size,
                              hipStream_t stream) {
  const float* pred2d = (const float*)d_in[0];  // (B,V,J,2)
  const float* gt3d   = (const float*)d_in[1];  // (B,J,3)
  const float* proj   = (const float*)d_in[2];  // (B,V,3,4)
  const float* valid  = (const float*)d_in[3];  // (B,V,J)
  float* out = (float*)d_out;

  zero_kernel<<<1, 32, 0, stream>>>(out);

  const int nprob = NB * NJ;                    // 524288, divisible by 256
  dlt_loss_kernel<<<nprob / THREADS, THREADS, 0, stream>>>(
      pred2d, gt3d, proj, valid, out);

  (void)in_sizes; (void)n_in; (void)out_size; (void)d_ws; (void)ws_size;
}